// MoE_53128745452100
// MI455X (gfx1250) — compile-verified
//
#include <hip/hip_runtime.h>

// ---------------- problem constants ----------------
#define T_TOK   8192          // 4 * 2048 tokens
#define EMB     1024
#define EXPD    2048
#define NE      8
#define CAP     8192          // max tokens per expert (top-k indices distinct)
#define MT      32            // token-tile (2 WMMA M-subtiles)
#define TPB     512           // 16 waves of 32
#define NWAVE   16
#define TILES_PER_E (CAP / MT)   // 256
#define RBLK    256           // block size for router / zero kernels

// padded LDS row strides (floats): stride % 64 == 4 -> 16 rows hit distinct banks
#define XSTR 1028             // X tile rows (K = 1024)
#define HSTR 2052             // H / G tile rows (K = 2048)

typedef float v2f __attribute__((ext_vector_type(2)));
typedef float v8f __attribute__((ext_vector_type(8)));

// D = A(16x4 f32) * B(4x16 f32) + C(16x16 f32), full fp32 precision.
__device__ __forceinline__ v8f wmma4_f32(v2f a, v2f b, v8f c) {
  return __builtin_amdgcn_wmma_f32_16x16x4_f32(
      /*neg_a=*/false, a, /*neg_b=*/false, b,
      /*c_mod=*/(short)0, c, /*reuse_a=*/false, /*reuse_b=*/false);
}

// ---------------- kernel 0: zero output + counters ----------------
__global__ void moe_zero(float* __restrict__ out, int n, int* __restrict__ counts) {
  int i = blockIdx.x * blockDim.x + threadIdx.x;
  if (i < n) out[i] = 0.0f;
  if (i < NE) counts[i] = 0;
}

// ---------------- kernel 1: router + top-2 + compaction ----------------
// one wave32 per token
__global__ void moe_router(const float* __restrict__ x,
                           const float* __restrict__ wr,
                           const float* __restrict__ br,
                           int* __restrict__ counts,
                           int* __restrict__ tok_list,
                           float* __restrict__ p_list) {
  const int lane = threadIdx.x & 31;
  const int wv   = threadIdx.x >> 5;
  const int t    = blockIdx.x * (RBLK / 32) + wv;
  if (t >= T_TOK) return;

  const float* xr = x + (size_t)t * EMB;
  float acc[NE];
#pragma unroll
  for (int e = 0; e < NE; ++e) acc[e] = 0.0f;

  for (int d = lane; d < EMB; d += 32) {
    const float xv = xr[d];
    const float* wrow = wr + (size_t)d * NE;
#pragma unroll
    for (int e = 0; e < NE; ++e) acc[e] += xv * wrow[e];
  }
#pragma unroll
  for (int off = 16; off >= 1; off >>= 1) {
#pragma unroll
    for (int e = 0; e < NE; ++e) acc[e] += __shfl_xor(acc[e], off, 32);
  }

  if (lane == 0) {
    float p[NE];
    float mx = -3.4e38f;
#pragma unroll
    for (int e = 0; e < NE; ++e) {
      p[e] = acc[e] + br[e];
      mx = fmaxf(mx, p[e]);
    }
    float den = 0.0f;
#pragma unroll
    for (int e = 0; e < NE; ++e) { p[e] = expf(p[e] - mx); den += p[e]; }
    const float inv = 1.0f / den;

    int i1 = 0; float p1 = p[0];
#pragma unroll
    for (int e = 1; e < NE; ++e) { if (p[e] > p1) { p1 = p[e]; i1 = e; } }
    int i2 = (i1 == 0) ? 1 : 0; float p2 = p[i2];
#pragma unroll
    for (int e = 0; e < NE; ++e) {
      if (e != i1 && p[e] > p2) { p2 = p[e]; i2 = e; }
    }
    p1 *= inv; p2 *= inv;

    int pos1 = atomicAdd(&counts[i1], 1);
    tok_list[i1 * CAP + pos1] = t;  p_list[i1 * CAP + pos1] = p1;
    int pos2 = atomicAdd(&counts[i2], 1);
    tok_list[i2 * CAP + pos2] = t;  p_list[i2 * CAP + pos2] = p2;
  }
}

// ---------------- kernel 2: fused routed expert MLP ----------------
// 16 waves; one 32-token tile of one expert, fully fused.
// One 256KB LDS arena holds X -> H -> G sequentially; each phase keeps its
// output in registers until a barrier, then overwrites the arena.
__global__ __launch_bounds__(TPB, 1)
void moe_expert(const float* __restrict__ x,
                const float* __restrict__ w1, const float* __restrict__ b1,
                const float* __restrict__ wg, const float* __restrict__ bg,
                const float* __restrict__ w2, const float* __restrict__ b2,
                const int*  __restrict__ counts,
                const int*  __restrict__ tok_list,
                const float* __restrict__ p_list,
                float* __restrict__ out) {
  extern __shared__ char smem[];
  float* AR    = (float*)smem;                            // 32 x HSTR floats arena
  int*   tok_s = (int*)(smem + (size_t)MT * HSTR * 4);
  float* p_s   = (float*)(tok_s + MT);

  const int e     = blockIdx.x / TILES_PER_E;
  const int tile  = blockIdx.x % TILES_PER_E;
  const int cnt   = counts[e];
  const int start = tile * MT;
  if (start >= cnt) return;
  const int valid = min(MT, cnt - start);

  const int tid = threadIdx.x;
  if (tid < MT) {
    const int slot = (tid < valid) ? (start + tid) : start;   // pad with row 0
    tok_s[tid] = tok_list[e * CAP + slot];
    p_s[tid]   = (tid < valid) ? p_list[e * CAP + slot] : 0.0f;
  }
  __syncthreads();

  // gather 32 token rows of x into the arena (coalesced), stride XSTR
  for (int idx = tid; idx < MT * EMB; idx += TPB) {
    const int m = idx >> 10;            // / EMB
    const int d = idx & (EMB - 1);
    AR[m * XSTR + d] = x[(size_t)tok_s[m] * EMB + d];
  }
  __syncthreads();

  const int lane = tid & 31;
  const int wv   = tid >> 5;
  const int nl   = lane & 15;           // N (and A-row) within a 16-wide tile
  const int hi   = lane >> 4;           // half-wave: K offset +2

  const float* w1e = w1 + (size_t)e * EMB  * EXPD;
  const float* wge = wg + (size_t)e * EXPD * EXPD;
  const float* w2e = w2 + (size_t)e * EXPD * EMB;

  v8f accA[8], accB[8];                 // [n-tile][M-subtile 0/1]

  // ---- phase 2: H = X(32x1024) @ w1e(1024x2048) + b1 ; H kept in regs ----
#pragma unroll
  for (int i = 0; i < 8; ++i) {
    const int n0 = (wv + i * NWAVE) * 16;
    const float* bp = w1e + (n0 + nl);
    const float* a0 = AR + nl * XSTR + 2 * hi;          // rows 0..15
    const float* a1 = AR + (16 + nl) * XSTR + 2 * hi;   // rows 16..31
    v8f c0 = {}, c1 = {};
#pragma unroll 4
    for (int k0 = 0; k0 < EMB; k0 += 4) {
      const int kk = k0 + 2 * hi;
      v2f av0 = *(const v2f*)(a0 + k0);
      v2f av1 = *(const v2f*)(a1 + k0);
      v2f b;                                            // shared by both subtiles
      b.x = bp[(size_t)kk * EXPD];
      b.y = bp[(size_t)(kk + 1) * EXPD];
      c0 = wmma4_f32(av0, b, c0);
      c1 = wmma4_f32(av1, b, c1);
    }
    accA[i] = c0; accB[i] = c1;
  }
  __syncthreads();                      // all X reads done
#pragma unroll
  for (int i = 0; i < 8; ++i) {         // write H(+bias) over the arena
    const int n0 = (wv + i * NWAVE) * 16;
    const float bias = b1[e * EXPD + n0 + nl];
#pragma unroll
    for (int r = 0; r < 8; ++r) {
      const int m = r + hi * 8;
      AR[m * HSTR + n0 + nl]        = accA[i][r] + bias;
      AR[(m + 16) * HSTR + n0 + nl] = accB[i][r] + bias;
    }
  }
  __syncthreads();

  // ---- phase 3: G = relu(H(32x2048) @ wge(2048x2048) + bg) ; G in regs ----
#pragma unroll
  for (int i = 0; i < 8; ++i) {
    const int n0 = (wv + i * NWAVE) * 16;
    const float* bp = wge + (n0 + nl);
    const float* a0 = AR + nl * HSTR + 2 * hi;
    const float* a1 = AR + (16 + nl) * HSTR + 2 * hi;
    v8f c0 = {}, c1 = {};
#pragma unroll 4
    for (int k0 = 0; k0 < EXPD; k0 += 4) {
      const int kk = k0 + 2 * hi;
      v2f av0 = *(const v2f*)(a0 + k0);
      v2f av1 = *(const v2f*)(a1 + k0);
      v2f b;
      b.x = bp[(size_t)kk * EXPD];
      b.y = bp[(size_t)(kk + 1) * EXPD];
      c0 = wmma4_f32(av0, b, c0);
      c1 = wmma4_f32(av1, b, c1);
    }
    accA[i] = c0; accB[i] = c1;
  }
  __syncthreads();                      // all H reads done
#pragma unroll
  for (int i = 0; i < 8; ++i) {         // write G = relu(acc+bg) over arena
    const int n0 = (wv + i * NWAVE) * 16;
    const float bias = bg[e * EXPD + n0 + nl];
#pragma unroll
    for (int r = 0; r < 8; ++r) {
      const int m = r + hi * 8;
      AR[m * HSTR + n0 + nl]        = fmaxf(accA[i][r] + bias, 0.0f);
      AR[(m + 16) * HSTR + n0 + nl] = fmaxf(accB[i][r] + bias, 0.0f);
    }
  }
  __syncthreads();

  // ---- phase 4: y = G(32x2048) @ w2e(2048x1024) + b2 ; out += p * y ----
#pragma unroll
  for (int i = 0; i < 4; ++i) {
    const int n0 = (wv + i * NWAVE) * 16;
    const float* bp = w2e + (n0 + nl);
    const float* a0 = AR + nl * HSTR + 2 * hi;
    const float* a1 = AR + (16 + nl) * HSTR + 2 * hi;
    v8f c0 = {}, c1 = {};
#pragma unroll 4
    for (int k0 = 0; k0 < EXPD; k0 += 4) {
      const int kk = k0 + 2 * hi;
      v2f av0 = *(const v2f*)(a0 + k0);
      v2f av1 = *(const v2f*)(a1 + k0);
      v2f b;
      b.x = bp[(size_t)kk * EMB];
      b.y = bp[(size_t)(kk + 1) * EMB];
      c0 = wmma4_f32(av0, b, c0);
      c1 = wmma4_f32(av1, b, c1);
    }
    const float bias = b2[e * EMB + n0 + nl];
#pragma unroll
    for (int r = 0; r < 8; ++r) {
      const int m = r + hi * 8;
      if (m < valid)
        atomicAdd(&out[(size_t)tok_s[m] * EMB + n0 + nl],
                  p_s[m] * (c0[r] + bias));
      if (m + 16 < valid)
        atomicAdd(&out[(size_t)tok_s[m + 16] * EMB + n0 + nl],
                  p_s[m + 16] * (c1[r] + bias));
    }
  }
}

// ---------------- host launcher ----------------
extern "C" void kernel_launch(void* const* d_in, const int* in_sizes, int n_in,
                              void* d_out, int out_size, void* d_ws, size_t ws_size,
                              hipStream_t stream) {
  const float* x  = (const float*)d_in[0];
  const float* wr = (const float*)d_in[1];
  const float* br = (const float*)d_in[2];
  const float* w1 = (const float*)d_in[3];
  const float* b1 = (const float*)d_in[4];
  const float* wg = (const float*)d_in[5];
  const float* bg = (const float*)d_in[6];
  const float* w2 = (const float*)d_in[7];
  const float* b2 = (const float*)d_in[8];
  float* out = (float*)d_out;

  char* ws = (char*)d_ws;
  int*   counts   = (int*)ws;                                   // 8 ints
  int*   tok_list = (int*)(ws + 256);                           // 8*8192 ints
  float* p_list   = (float*)(ws + 256 + (size_t)NE * CAP * 4);  // 8*8192 floats

  const int n = T_TOK * EMB;
  moe_zero<<<(n + RBLK - 1) / RBLK, RBLK, 0, stream>>>(out, n, counts);

  moe_router<<<T_TOK / (RBLK / 32), RBLK, 0, stream>>>(x, wr, br,
                                                       counts, tok_list, p_list);

  const size_t lds_bytes =
      (size_t)MT * HSTR * sizeof(float) + MT * (sizeof(int) + sizeof(float));
  moe_expert<<<NE * TILES_PER_E, TPB, lds_bytes, stream>>>(
      x, w1, b1, wg, bg, w2, b2, counts, tok_list, p_list, out);
}